// SinkhornRouter_44590350467593
// MI455X (gfx1250) — compile-verified
//
#include <hip/hip_runtime.h>
#include <stdint.h>

typedef __attribute__((ext_vector_type(2))) float v2f;
typedef __attribute__((ext_vector_type(8))) float v8f;

#define BATCH   4
#define NTOK    1024
#define NPOS    8192
#define DIM     1024
#define ITILES  8
#define ICHUNK  (NTOK / ITILES)   // 128
#define INV_T   (1.0f / 0.7f)

// ---------------------------------------------------------------------------
// Streaming (online) logsumexp update, branchless so wave32 lanes stay coherent
// ---------------------------------------------------------------------------
__device__ __forceinline__ void lse_push(float v, float& m, float& s) {
  float nm = fmaxf(m, v);
  s = s * __expf(m - nm) + __expf(v - nm);
  m = nm;
}

// ---------------------------------------------------------------------------
// Kernel 1: scores[b,j] = dot(x[b,j,:], routing_token) / TEMPERATURE
// One wave per 16 rows using V_WMMA_F32_16X16X4_F32.
//   A (16x4 f32):  lane L (row = L&15): VGPR0 = x[row, k + 2*(L>>4)],
//                                       VGPR1 = x[row, k + 2*(L>>4) + 1]
//   B (4x16 f32):  broadcast routing_token chunk across all 16 columns,
//                  lanes 0-15 hold K=0/1, lanes 16-31 hold K=2/3
//   D (16x16 f32): every column identical; lane 0 VGPRs 0..7 = rows 0..7,
//                  lane 16 VGPRs 0..7 = rows 8..15
// x is streamed with non-temporal loads so it does not evict gumbel from L2.
// ---------------------------------------------------------------------------
__global__ __launch_bounds__(256) void gemv_scores(
    const float* __restrict__ x, const float* __restrict__ rt,
    float* __restrict__ sT) {
  const int lane = threadIdx.x & 31;
  const int m    = lane & 15;
  const int half = lane >> 4;
  const size_t wave = (size_t)((blockIdx.x * blockDim.x + threadIdx.x) >> 5);
  const size_t row0 = wave * 16;                 // 16 rows of (b*n) = 32768
  const float4* rt4 = (const float4*)rt;
  const float* xr = x + (row0 + (size_t)m) * DIM + 2 * half;

#if defined(__gfx1250__) && __has_builtin(__builtin_amdgcn_wmma_f32_16x16x4_f32)
  v8f acc = {};
  for (int k = 0; k < DIM; k += 4) {
    v2f a = __builtin_nontemporal_load((const v2f*)(xr + k));
    float4 r = rt4[k >> 2];                      // uniform -> s_load_b128
    v2f bv;
    bv.x = half ? r.z : r.x;                     // K = 0 or 2
    bv.y = half ? r.w : r.y;                     // K = 1 or 3
    acc = __builtin_amdgcn_wmma_f32_16x16x4_f32(
        false, a, false, bv, (short)0, acc, false, false);
  }
  if (m == 0) {                                  // lane 0 -> rows 0..7, lane 16 -> rows 8..15
#pragma unroll
    for (int rr = 0; rr < 8; ++rr)
      sT[row0 + (size_t)(half * 8 + rr)] = acc[rr] * INV_T;
  }
#else
  float partial = 0.f;
  for (int k = 0; k < DIM; k += 4) {
    v2f a = __builtin_nontemporal_load((const v2f*)(xr + k));
    float4 r = rt4[k >> 2];
    partial += a.x * (half ? r.z : r.x) + a.y * (half ? r.w : r.y);
  }
  partial += __shfl_xor(partial, 16, 32);
  if (half == 0) sT[row0 + (size_t)m] = partial * INV_T;
#endif
}

// ---------------------------------------------------------------------------
// Kernel 2: partial column LSE:  pm/ps[b,itile,j] = online-LSE over an i-chunk
// of (sT[b,j] + g[b,i,j] - R[b,i]).  Each thread owns 4 contiguous j.
// ---------------------------------------------------------------------------
__global__ __launch_bounds__(256) void col_pass(
    const float* __restrict__ g, const float* __restrict__ sT,
    const float* __restrict__ R, float* __restrict__ pm,
    float* __restrict__ ps) {
  const int jtile = blockIdx.x, itile = blockIdx.y, b = blockIdx.z;
  const int j = jtile * 1024 + threadIdx.x * 4;
  const float4 sv = *(const float4*)(sT + (size_t)b * NPOS + j);
  const float* Rb = R + b * NTOK + itile * ICHUNK;
  const float* gbase =
      g + ((size_t)(b * NTOK + itile * ICHUNK)) * NPOS + j;

  float m0 = -INFINITY, m1 = -INFINITY, m2 = -INFINITY, m3 = -INFINITY;
  float s0 = 0.f, s1 = 0.f, s2 = 0.f, s3 = 0.f;
  for (int ii = 0; ii < ICHUNK; ++ii) {
    float4 g4 = *(const float4*)(gbase + (size_t)ii * NPOS);
    if (ii + 4 < ICHUNK)
      __builtin_prefetch(gbase + (size_t)(ii + 4) * NPOS, 0, 3);  // global_prefetch_b8
    float Ri = Rb[ii];                                            // uniform
    lse_push(sv.x + g4.x - Ri, m0, s0);
    lse_push(sv.y + g4.y - Ri, m1, s1);
    lse_push(sv.z + g4.z - Ri, m2, s2);
    lse_push(sv.w + g4.w - Ri, m3, s3);
  }
  const size_t o = ((size_t)(b * ITILES + itile)) * NPOS + j;
  *(float4*)(pm + o) = make_float4(m0, m1, m2, m3);
  *(float4*)(ps + o) = make_float4(s0, s1, s2, s3);
}

// ---------------------------------------------------------------------------
// Kernel 3: combine ITILES partial LSEs -> C[b,j] = lse_i(sT[j]+g[i,j]-R[i])
// ---------------------------------------------------------------------------
__global__ __launch_bounds__(256) void combine_c(
    const float* __restrict__ pm, const float* __restrict__ ps,
    float* __restrict__ C) {
  const int gid = blockIdx.x * blockDim.x + threadIdx.x;  // 0 .. B*NPOS-1
  const int b = gid >> 13;
  const int j = gid & (NPOS - 1);
  float m = -INFINITY;
#pragma unroll
  for (int c = 0; c < ITILES; ++c)
    m = fmaxf(m, pm[((size_t)(b * ITILES + c)) * NPOS + j]);
  float s = 0.f;
#pragma unroll
  for (int c = 0; c < ITILES; ++c) {
    const size_t o = ((size_t)(b * ITILES + c)) * NPOS + j;
    s += ps[o] * __expf(pm[o] - m);
  }
  C[gid] = m + __logf(s);
}

// ---------------------------------------------------------------------------
// Kernel 4: row LSE:  R[b,i] = lse_j(sT[j] + g[b,i,j] - C[b,j])
// One block per (b,i) row of 8192; LDS tree combines per-thread (m,s).
// ---------------------------------------------------------------------------
__global__ __launch_bounds__(256) void row_pass(
    const float* __restrict__ g, const float* __restrict__ sT,
    const float* __restrict__ C, float* __restrict__ R) {
  const int row = blockIdx.x;           // b*NTOK + i
  const int b = row >> 10;
  const float* grow = g + (size_t)row * NPOS;
  const float* sTb = sT + (size_t)b * NPOS;
  const float* Cb  = C  + (size_t)b * NPOS;

  float m = -INFINITY, s = 0.f;
  for (int c = 0; c < NPOS / 1024; ++c) {
    const int j = c * 1024 + threadIdx.x * 4;
    float4 g4 = *(const float4*)(grow + j);
    float4 sv = *(const float4*)(sTb + j);
    float4 cv = *(const float4*)(Cb + j);
    lse_push(sv.x + g4.x - cv.x, m, s);
    lse_push(sv.y + g4.y - cv.y, m, s);
    lse_push(sv.z + g4.z - cv.z, m, s);
    lse_push(sv.w + g4.w - cv.w, m, s);
  }
  __shared__ float sm[256], ss[256];
  sm[threadIdx.x] = m; ss[threadIdx.x] = s;
  __syncthreads();
  for (int off = 128; off > 0; off >>= 1) {
    if (threadIdx.x < off) {
      float m2 = sm[threadIdx.x + off], s2 = ss[threadIdx.x + off];
      float M = fmaxf(sm[threadIdx.x], m2);
      ss[threadIdx.x] = ss[threadIdx.x] * __expf(sm[threadIdx.x] - M) +
                        s2 * __expf(m2 - M);
      sm[threadIdx.x] = M;
    }
    __syncthreads();
  }
  if (threadIdx.x == 0) R[row] = sm[0] + __logf(ss[0]);
}

// ---------------------------------------------------------------------------
// Kernel 5: per (b,i) row: argmax/max of (sT[j]+g[i,j]-C[j]); exp applied to
// the single max (monotone); scores = s + (1 - s), masked by mask[b, idx].
// ---------------------------------------------------------------------------
__global__ __launch_bounds__(256) void final_pass(
    const float* __restrict__ g, const float* __restrict__ sT,
    const float* __restrict__ C, const float* __restrict__ R,
    const unsigned char* __restrict__ mask, float* __restrict__ outS,
    float* __restrict__ outI) {
  const int row = blockIdx.x;           // b*NTOK + i
  const int b = row >> 10;
  const float* grow = g + (size_t)row * NPOS;
  const float* sTb = sT + (size_t)b * NPOS;
  const float* Cb  = C  + (size_t)b * NPOS;

  float best = -INFINITY;
  int bidx = 0x7fffffff;
  for (int c = 0; c < NPOS / 1024; ++c) {
    const int j = c * 1024 + threadIdx.x * 4;
    float4 g4 = *(const float4*)(grow + j);
    float4 sv = *(const float4*)(sTb + j);
    float4 cv = *(const float4*)(Cb + j);
    float v[4] = {sv.x + g4.x - cv.x, sv.y + g4.y - cv.y,
                  sv.z + g4.z - cv.z, sv.w + g4.w - cv.w};
#pragma unroll
    for (int q = 0; q < 4; ++q) {
      const int jj = j + q;
      if (v[q] > best || (v[q] == best && jj < bidx)) { best = v[q]; bidx = jj; }
    }
  }
  __shared__ float svv[256];
  __shared__ int sii[256];
  svv[threadIdx.x] = best; sii[threadIdx.x] = bidx;
  __syncthreads();
  for (int off = 128; off > 0; off >>= 1) {
    if (threadIdx.x < off) {
      float v2 = svv[threadIdx.x + off]; int i2 = sii[threadIdx.x + off];
      if (v2 > svv[threadIdx.x] ||
          (v2 == svv[threadIdx.x] && i2 < sii[threadIdx.x])) {
        svv[threadIdx.x] = v2; sii[threadIdx.x] = i2;
      }
    }
    __syncthreads();
  }
  if (threadIdx.x == 0) {
    float sfin = __expf(svv[0] - R[row]);
    float score = sfin + (1.0f - sfin);             // matches reference fp order
    unsigned char mk = mask[(size_t)b * NPOS + sii[0]];
    outS[row] = mk ? score : 0.0f;
    outI[row] = (float)sii[0];
  }
}

// ---------------------------------------------------------------------------
extern "C" void kernel_launch(void* const* d_in, const int* in_sizes, int n_in,
                              void* d_out, int out_size, void* d_ws,
                              size_t ws_size, hipStream_t stream) {
  (void)in_sizes; (void)n_in; (void)out_size; (void)ws_size;
  const float* x  = (const float*)d_in[0];
  const float* rt = (const float*)d_in[1];
  const float* g  = (const float*)d_in[2];
  const unsigned char* mask = (const unsigned char*)d_in[3];
  float* out = (float*)d_out;

  float* ws = (float*)d_ws;
  float* sT = ws;             // B*NPOS          = 32768 floats
  float* R  = ws + 32768;     // B*NTOK          =  4096
  float* C  = ws + 36864;     // B*NPOS          = 32768
  float* pm = ws + 69632;     // B*ITILES*NPOS   = 262144
  float* ps = ws + 331776;    // B*ITILES*NPOS   = 262144   (total ~2.3 MB)

  hipMemsetAsync(R, 0, BATCH * NTOK * sizeof(float), stream);

  // scores/T via WMMA: 2048 waves, 16 rows each
  gemv_scores<<<(BATCH * NPOS) / 128, 256, 0, stream>>>(x, rt, sT);

  for (int it = 0; it < 8; ++it) {
    col_pass<<<dim3(NPOS / 1024, ITILES, BATCH), 256, 0, stream>>>(g, sT, R, pm, ps);
    combine_c<<<(BATCH * NPOS) / 256, 256, 0, stream>>>(pm, ps, C);
    row_pass<<<BATCH * NTOK, 256, 0, stream>>>(g, sT, C, R);
  }
  final_pass<<<BATCH * NTOK, 256, 0, stream>>>(g, sT, C, R, mask, out,
                                               out + BATCH * NTOK);
}